// BilinearInterpolation3D_59554016526641
// MI455X (gfx1250) — compile-verified
//
#include <hip/hip_runtime.h>
#include <stdint.h>

// Problem dims (fixed by the reference: x = (2,1,128,128,128) f32, scale 2)
#define Bq   2
#define Dq   128
#define Hq   128
#define Wq   128
#define HOq  256
#define WOq  256
#define DOq  256
#define TROWS 6    // input rows staged per block (covers 8 output rows + halo)
#define OROWS 8    // output rows per block

typedef __attribute__((ext_vector_type(4))) float        f4;
typedef __attribute__((ext_vector_type(4))) unsigned int tdm_v4u;
typedef __attribute__((ext_vector_type(8))) int          tdm_v8i;
typedef __attribute__((ext_vector_type(4))) int          tdm_v4i;

#if __has_builtin(__builtin_amdgcn_tensor_load_to_lds) && __has_builtin(__builtin_amdgcn_s_wait_tensorcnt)
#define STAGE_TDM 1
#elif __has_builtin(__builtin_amdgcn_global_load_async_to_lds_b128) && __has_builtin(__builtin_amdgcn_s_wait_asynccnt)
#define STAGE_ASYNC 1
#endif

__global__ __launch_bounds__(256) void bilin3d_up2_kernel(const float* __restrict__ x,
                                                          float* __restrict__ out) {
  __shared__ float tile[TROWS * Wq];   // 3 KB staged input rows

  const int hb = blockIdx.x;           // 0..31  (output-row tile)
  const int d  = blockIdx.y;           // 0..127 (input depth slice)
  const int b  = blockIdx.z;           // 0..1
  const int r0 = hb * OROWS;           // first output row of this tile

  // Input rows needed: [r0/2 - 1, r0/2 + 4]; clamp the window so the staged
  // tile is always fully inside the tensor (border taps are handled by
  // clamping indices into the window; constant 0.25/0.75 weights stay exact
  // because both taps collapse to the same element at the borders).
  int hbase = (r0 >> 1) - 1;
  hbase = hbase < 0 ? 0 : hbase;
  hbase = hbase > (Hq - TROWS) ? (Hq - TROWS) : hbase;

  const float* gsrc = x + ((size_t)((b * Dq + d) * Hq + hbase)) * Wq; // 6 contiguous rows

#if defined(STAGE_TDM)
  // ---- CDNA5 Tensor Data Mover: one D# moves the whole 6x128 f32 tile ----
  if (threadIdx.x == 0) {
    unsigned long long ga = (unsigned long long)(const void*)gsrc;
    unsigned lds_off =
        (unsigned)(unsigned long long)(__attribute__((address_space(3))) float*)&tile[0];

    tdm_v4u g0;
    g0.x = 1u;                                            // count=1 (valid D#)
    g0.y = lds_off;                                       // lds_addr [63:32]
    g0.z = (unsigned)(ga & 0xffffffffull);                // global_addr[31:0]
    g0.w = (unsigned)((ga >> 32) & 0x01ffffffull)         // global_addr[56:32]
         | (2u << 30);                                    // type=2 ("image")

    tdm_v8i g1;
    g1[0] = (int)(2u << 16);                              // data_size=2 -> 4 bytes
    g1[1] = (int)((unsigned)Wq << 16);                    // tensor_dim0[15:0]=128 @ bits[63:48]
    g1[2] = (int)((unsigned)(Hq - hbase) << 16);          // tensor_dim1[15:0] (rows remaining >= 6)
    g1[3] = (int)((unsigned)Wq << 16);                    // tile_dim0 = 128 @ bits[127:112]
    g1[4] = TROWS;                                        // tile_dim1 = 6, tile_dim2 = 0
    g1[5] = Wq;                                           // tensor_dim0_stride[31:0] = 128
    g1[6] = 0;                                            // stride hi / dim1_stride (unused, 2D)
    g1[7] = 0;
    tdm_v4i gz4 = {0, 0, 0, 0};
    tdm_v8i gz8 = {0, 0, 0, 0, 0, 0, 0, 0};

    // amdgpu-toolchain (clang-23) 6-arg form:
    // (uint32x4 g0, int32x8 g1, int32x4 g2, int32x4 g3, int32x8 g4, i32 cpol)
    __builtin_amdgcn_tensor_load_to_lds(g0, g1, gz4, gz4, gz8, 0);
    __builtin_amdgcn_s_wait_tensorcnt(0);                 // s_wait_tensorcnt 0
  }
#elif defined(STAGE_ASYNC)
  {
    const int t = threadIdx.x;
    if (t < (TROWS * Wq) / 4) {                           // 192 lanes, one b128 each
      __builtin_amdgcn_global_load_async_to_lds_b128(
          (const __attribute__((address_space(1))) void*)(gsrc + t * 4),
          (__attribute__((address_space(3))) void*)&tile[t * 4], 0, 0);
    }
    __builtin_amdgcn_s_wait_asynccnt(0);
  }
#else
  {
    const int t = threadIdx.x;
    if (t < (TROWS * Wq) / 4) {
      f4 v = *(const f4*)(gsrc + t * 4);
      *(f4*)&tile[t * 4] = v;
    }
  }
#endif
  __syncthreads();

  // ---- compute: constant-weight separable 2x bilinear from LDS ----
  const int tid = threadIdx.x;
  const int r   = tid >> 5;        // output row within tile: 0..7
  const int c4a = tid & 31;        // float4 column index; second job at +32

  const int ho = r0 + r;
  const int k  = ho >> 1;
  int hi0, hi1;
  float hw0, hw1;
  if (ho & 1) { hi0 = k;                         hi1 = (k + 1 < Hq) ? k + 1 : Hq - 1;
                hw0 = 0.75f; hw1 = 0.25f; }
  else        { hi0 = (k - 1 > 0) ? k - 1 : 0;   hi1 = k;
                hw0 = 0.25f; hw1 = 0.75f; }

  const float* row0 = &tile[(hi0 - hbase) * Wq];
  const float* row1 = &tile[(hi1 - hbase) * Wq];

  float* o0 = out + (((size_t)(b * DOq + 2 * d)) * HOq + ho) * WOq; // D slice 2d
  float* o1 = o0 + (size_t)HOq * WOq;                               // D slice 2d+1 (nearest repeat)

#pragma unroll
  for (int half = 0; half < 2; ++half) {
    const int c   = c4a + half * 32;   // float4 index 0..63
    const int wo0 = c * 4;
    const int ib  = 2 * c;
    const int ia  = (ib - 1 > 0) ? ib - 1 : 0;
    const int ic  = ib + 1;
    const int id  = (ib + 2 < Wq) ? ib + 2 : Wq - 1;

    // H-direction interpolation of the 4 needed input columns
    const float ta = hw0 * row0[ia] + hw1 * row1[ia];
    const float tb = hw0 * row0[ib] + hw1 * row1[ib];
    const float tc = hw0 * row0[ic] + hw1 * row1[ic];
    const float td = hw0 * row0[id] + hw1 * row1[id];

    // W-direction: fixed 0.25/0.75 pattern by output parity
    f4 acc;
    acc[0] = 0.25f * ta + 0.75f * tb;   // wo = 4c   (even)
    acc[1] = 0.75f * tb + 0.25f * tc;   // wo = 4c+1 (odd)
    acc[2] = 0.25f * tb + 0.75f * tc;   // wo = 4c+2 (even)
    acc[3] = 0.75f * tc + 0.25f * td;   // wo = 4c+3 (odd)

    // streaming (non-temporal) 128-bit stores; output never re-read
    __builtin_nontemporal_store(acc, (f4*)(o0 + wo0));
    __builtin_nontemporal_store(acc, (f4*)(o1 + wo0));
  }
}

extern "C" void kernel_launch(void* const* d_in, const int* in_sizes, int n_in,
                              void* d_out, int out_size, void* d_ws, size_t ws_size,
                              hipStream_t stream) {
  (void)in_sizes; (void)n_in; (void)out_size; (void)d_ws; (void)ws_size;
  const float* x = (const float*)d_in[0];
  float* out = (float*)d_out;

  dim3 grid(HOq / OROWS, Dq, Bq);   // (32, 128, 2) = 8192 blocks
  bilin3d_up2_kernel<<<grid, 256, 0, stream>>>(x, out);
}